// IQARegression_27908697489631
// MI455X (gfx1250) — compile-verified
//
#include <hip/hip_runtime.h>
#include <math.h>

typedef float v2f __attribute__((ext_vector_type(2)));
typedef float v8f __attribute__((ext_vector_type(8)));
typedef unsigned int u32x4 __attribute__((ext_vector_type(4)));
typedef int i32x8 __attribute__((ext_vector_type(8)));
typedef int i32x4 __attribute__((ext_vector_type(4)));

#define B_TOK 32768
#define D_DIM 1024
#define SEL_N 512
#define E_N 6
#define SH_N 32
#define INNER_N 64
#define HID_N 128

// ---- workspace layout (in floats) ----
#define W1T_OFF 0
#define W1T_SZ (E_N * D_DIM * HID_N)       // 786432
#define B1T_OFF (W1T_OFF + W1T_SZ)
#define B1T_SZ (E_N * HID_N)               // 768
#define WCOMB_OFF (B1T_OFF + B1T_SZ)
#define WCOMB_SZ (SEL_N * SH_N)            // 16384
#define BCOMB_OFF (WCOMB_OFF + WCOMB_SZ)
#define BCOMB_SZ SH_N
#define RANK_OFF (BCOMB_OFF + BCOMB_SZ)    // int region
#define RANK_SZ (E_N * D_DIM)

__device__ __forceinline__ v8f wmma4(v2f a, v2f b, v8f c) {
  // D(16x16,f32) = A(16x4,f32) x B(4x16,f32) + C
  return __builtin_amdgcn_wmma_f32_16x16x4_f32(false, a, false, b, (short)0, c,
                                               false, false);
}

// ---- CDNA5 async global->LDS copies (ASYNCcnt) ----
__device__ __forceinline__ void async_cp_b128(void* lds, const void* g) {
  asm volatile("global_load_async_to_lds_b128 %0, %1, off" ::"v"(
                   (unsigned)(unsigned long long)lds),
               "v"(g)
               : "memory");
}
__device__ __forceinline__ void async_cp_b32(void* lds, const void* g) {
  asm volatile("global_load_async_to_lds_b32 %0, %1, off" ::"v"(
                   (unsigned)(unsigned long long)lds),
               "v"(g)
               : "memory");
}
__device__ __forceinline__ void async_wait0() {
  asm volatile("s_wait_asynccnt 0x0" ::: "memory");
}

// ---- CDNA5 Tensor Data Mover: 2D tile (128 rows x 128 f32), memory row
// stride 1024 f32, LDS padded to 132-f32 rows via pad_interval=128DW/amount=4DW
__device__ __forceinline__ void tdm_load_w128x128(const float* src,
                                                  float* ldsdst) {
  const unsigned long long ga = (unsigned long long)src;
  const unsigned lds = (unsigned)(unsigned long long)ldsdst;
  u32x4 g0;
  g0[0] = 0x1u;                                   // count=1 (valid descriptor)
  g0[1] = lds;                                    // lds_addr (bytes)
  g0[2] = (unsigned)ga;                           // global_addr[31:0]
  g0[3] = (unsigned)(ga >> 32) | 0x80000000u;     // addr[56:32] | type=2
  i32x8 g1;
  g1[0] = (2 << 16)      // data_size = 4B
          | (1 << 20)    // pad_enable
          | (6 << 22)    // pad_interval: 2^(6+1)=128 DWORDs
          | (3 << 25);   // pad_amount: 3+1 = 4 DWORDs
  g1[1] = (HID_N & 0xFFFF) << 16;                 // tensor_dim0[15:0]=128
  g1[2] = (HID_N & 0xFFFF) << 16;                 // tensor_dim1[15:0]=128
  g1[3] = (HID_N & 0xFFFF) << 16;                 // tile_dim0=128
  g1[4] = HID_N;                                  // tile_dim1=128, tile_dim2=0
  g1[5] = D_DIM;                                  // tensor_dim0_stride=1024
  g1[6] = 0;
  g1[7] = 0;
  const i32x4 gz4 = {0, 0, 0, 0};
  const i32x8 gz8 = {0, 0, 0, 0, 0, 0, 0, 0};
  __builtin_amdgcn_tensor_load_to_lds(g0, g1, gz4, gz4, gz8, 0);
}
__device__ __forceinline__ void tdm_wait0() {
  __builtin_amdgcn_s_wait_tensorcnt(0);
}

// ---------------------------------------------------------------------------
// Setup 1: Wcomb = Wv @ (Wo @ Wsp), bcomb = bo@Wsp + bsp, top-k ranks per
// expert with exact jax.lax.top_k tie rules (desc value, ties -> lower index).
// ---------------------------------------------------------------------------
__global__ __launch_bounds__(1024) void moe_setup1(
    const float* __restrict__ Wv, const float* __restrict__ Wo,
    const float* __restrict__ bo, const float* __restrict__ Wsp,
    const float* __restrict__ bsp, const float* __restrict__ ml,
    float* __restrict__ ws) {
  __shared__ float sWos[INNER_N * SH_N];  // 64x32
  __shared__ float srow[D_DIM];
  const int t = threadIdx.x;
  float* wcomb = ws + WCOMB_OFF;
  float* bcomb = ws + BCOMB_OFF;
  int* rank = (int*)(ws + RANK_OFF);

  for (int idx = t; idx < INNER_N * SH_N; idx += 1024) {
    const int i = idx >> 5, j = idx & 31;
    float s = 0.f;
    for (int k = 0; k < 512; ++k) s = fmaf(Wo[i * 512 + k], Wsp[k * 32 + j], s);
    sWos[idx] = s;
  }
  if (t < SH_N) {
    float s = bsp[t];
    for (int k = 0; k < 512; ++k) s = fmaf(bo[k], Wsp[k * 32 + t], s);
    bcomb[t] = s;
  }
  __syncthreads();
  for (int idx = t; idx < SEL_N * SH_N; idx += 1024) {
    const int d = idx >> 5, j = idx & 31;
    float s = 0.f;
#pragma unroll
    for (int i = 0; i < INNER_N; ++i)
      s = fmaf(Wv[d * 64 + i], sWos[i * 32 + j], s);
    wcomb[idx] = s;
  }
  for (int e = 0; e < E_N; ++e) {
    __syncthreads();
    srow[t] = ml[e * D_DIM + t];
    __syncthreads();
    const float mv = srow[t];
    int r = 0;
    for (int d2 = 0; d2 < D_DIM; ++d2) {
      const float o = srow[d2];
      r += (o > mv) ? 1 : ((o == mv && d2 < t) ? 1 : 0);
    }
    rank[e * D_DIM + t] = r;
  }
}

// ---------------------------------------------------------------------------
// Setup 2: build folded weight W1t stored [e][n][d] and folded bias b1t[e][n].
// ---------------------------------------------------------------------------
__global__ __launch_bounds__(128) void moe_setup2(
    const float* __restrict__ W1, const float* __restrict__ b1,
    float* __restrict__ ws) {
  const int e = blockIdx.x >> 10;
  const int d = blockIdx.x & 1023;
  const int n = threadIdx.x;
  const float* wcomb = ws + WCOMB_OFF;
  const float* bcomb = ws + BCOMB_OFF;
  const int* rank = (const int*)(ws + RANK_OFF);
  float* w1t = ws + W1T_OFF;
  float* b1t = ws + B1T_OFF;

  const float* W1e = W1 + (size_t)e * (SEL_N + SH_N) * HID_N;
  float v = 0.f;
  const int r = rank[e * D_DIM + d];
  if (r < SEL_N) v = W1e[r * HID_N + n];
  if (d >= SEL_N) {
    const float* wc = wcomb + (d - SEL_N) * SH_N;
#pragma unroll
    for (int j = 0; j < SH_N; ++j)
      v = fmaf(wc[j], W1e[(SEL_N + j) * HID_N + n], v);
  }
  w1t[(size_t)(e * HID_N + n) * D_DIM + d] = v;
  if (d == 0) {
    float bb = b1[e * HID_N + n];
#pragma unroll
    for (int j = 0; j < SH_N; ++j)
      bb = fmaf(bcomb[j], W1e[(SEL_N + j) * HID_N + n], bb);
    b1t[e * HID_N + n] = bb;
  }
}

// ---------------------------------------------------------------------------
// Main fused kernel: double-buffered TDM/async staging pipelined with f32 WMMA.
// ---------------------------------------------------------------------------
#define MT 64
#define KC 128
#define XS 132  // padded LDS row stride (conflict-free ds_load_b64)

__global__ __launch_bounds__(512) void moe_main(
    const float* __restrict__ x, const float* __restrict__ Wg,
    const float* __restrict__ bg, const float* __restrict__ W2,
    const float* __restrict__ b2, const float* __restrict__ ws,
    float* __restrict__ out) {
  __shared__ float Xs[2][MT * XS];       // 2 x 64 x 132
  __shared__ float Wst[2][HID_N * XS];   // 2 x 128 x 132 (expert chunk [n][k])
  __shared__ float Wgs[2][KC * E_N];     // gate weight chunks
  __shared__ float gl[MT * E_N];         // gate logits
  __shared__ float eo[MT * E_N];         // expert outputs (pre-bias)

  const float* w1t = ws + W1T_OFF;
  const float* b1t = ws + B1T_OFF;

  const int t = threadIdx.x;
  const int wave = t >> 5;
  const int lane = t & 31;
  const int lh = lane >> 4;
  const int lm = lane & 15;
  const int mt = wave & 3;   // M tile (16 tokens)
  const int ng = wave >> 2;  // N group (32 cols)
  const int b0 = blockIdx.x * MT;

  v8f acc[E_N][2];
#pragma unroll
  for (int e = 0; e < E_N; ++e) {
    acc[e][0] = (v8f){0.f, 0.f, 0.f, 0.f, 0.f, 0.f, 0.f, 0.f};
    acc[e][1] = (v8f){0.f, 0.f, 0.f, 0.f, 0.f, 0.f, 0.f, 0.f};
  }

  const float* xblk = x + (size_t)b0 * D_DIM;
  float gacc = 0.f;
  const int gm = t & 63, ge = t >> 6;  // valid for t < 384

  const int arow = (mt * 16 + lm) * XS + 2 * lh;
  const int brow0 = (ng * 32 + lm) * XS + 2 * lh;
  const int brow1 = brow0 + 16 * XS;

  // per-thread staging coordinates
  const int xm = t >> 5;            // X chunk: 4 async b128 per thread
  const int xk = (t << 2) & 127;

  // ---- prologue: stage (kc=0, e=0) ----
#pragma unroll
  for (int j = 0; j < 4; ++j)
    async_cp_b128(&Xs[0][(xm + j * 16) * XS + xk],
                  xblk + (size_t)(xm + j * 16) * D_DIM + xk);
  for (int i = t; i < KC * E_N; i += 512)
    async_cp_b32(&Wgs[0][i], Wg + i);
  if (wave == 0) tdm_load_w128x128(w1t, &Wst[0][0]);

  int wbuf = 0, xbuf = 0, gbuf = 0;

  for (int kc = 0; kc < D_DIM; kc += KC) {
#pragma unroll
    for (int e = 0; e < E_N; ++e) {
      if (wave == 0) tdm_wait0();   // Wst[wbuf] DMA complete (wave 0 issued)
      async_wait0();                // my Xs/Wgs async copies complete
      __syncthreads();              // stage ready + prev-stage readers done

      // issue next stage into the other buffers (overlaps with WMMA below)
      if (e < E_N - 1) {
        if (wave == 0)
          tdm_load_w128x128(w1t + (size_t)(e + 1) * HID_N * D_DIM + kc,
                            &Wst[wbuf ^ 1][0]);
      } else if (kc + KC < D_DIM) {
        if (wave == 0)
          tdm_load_w128x128(w1t + (kc + KC), &Wst[wbuf ^ 1][0]);
#pragma unroll
        for (int j = 0; j < 4; ++j)
          async_cp_b128(&Xs[xbuf ^ 1][(xm + j * 16) * XS + xk],
                        xblk + (size_t)(xm + j * 16) * D_DIM + (kc + KC) + xk);
        for (int i = t; i < KC * E_N; i += 512)
          async_cp_b32(&Wgs[gbuf ^ 1][i], Wg + (kc + KC) * E_N + i);
      }

      // gate partial logits, once per K chunk
      if (e == 0 && t < MT * E_N) {
        float s = gacc;
#pragma unroll 8
        for (int k = 0; k < KC; ++k)
          s = fmaf(Xs[xbuf][gm * XS + k], Wgs[gbuf][k * E_N + ge], s);
        gacc = s;
      }

      // WMMA: 4 independent accumulation chains
      v8f c0 = acc[e][0], c1 = acc[e][1];
      v8f d0 = (v8f){0.f, 0.f, 0.f, 0.f, 0.f, 0.f, 0.f, 0.f};
      v8f d1 = (v8f){0.f, 0.f, 0.f, 0.f, 0.f, 0.f, 0.f, 0.f};
#pragma unroll 2
      for (int kk = 0; kk < KC; kk += 8) {
        const v2f a0 = *(const v2f*)&Xs[xbuf][arow + kk];
        const v2f a1 = *(const v2f*)&Xs[xbuf][arow + kk + 4];
        const v2f b00 = *(const v2f*)&Wst[wbuf][brow0 + kk];
        const v2f b01 = *(const v2f*)&Wst[wbuf][brow0 + kk + 4];
        const v2f b10 = *(const v2f*)&Wst[wbuf][brow1 + kk];
        const v2f b11 = *(const v2f*)&Wst[wbuf][brow1 + kk + 4];
        c0 = wmma4(a0, b00, c0);
        c1 = wmma4(a0, b10, c1);
        d0 = wmma4(a1, b01, d0);
        d1 = wmma4(a1, b11, d1);
      }
      acc[e][0] = c0 + d0;
      acc[e][1] = c1 + d1;

      wbuf ^= 1;
      if (e == E_N - 1) {
        xbuf ^= 1;
        gbuf ^= 1;
      }
    }
  }

  __syncthreads();
  if (t < MT * E_N) {
    gl[gm * E_N + ge] = gacc + bg[ge];
    eo[t] = 0.f;
  }
  __syncthreads();

  // epilogue: bias + ReLU + dot(W2), 16-lane butterfly reduce, LDS accumulate
#pragma unroll
  for (int e = 0; e < E_N; ++e) {
#pragma unroll
    for (int tt = 0; tt < 2; ++tt) {
      const int n = ng * 32 + tt * 16 + lm;
      const float bias = b1t[e * HID_N + n];
      const float w2v = W2[e * HID_N + n];
      float r[8];
#pragma unroll
      for (int v = 0; v < 8; ++v) {
        float h = acc[e][tt][v] + bias;
        h = h > 0.f ? h : 0.f;
        r[v] = h * w2v;
      }
#pragma unroll
      for (int off = 1; off < 16; off <<= 1) {
#pragma unroll
        for (int v = 0; v < 8; ++v) r[v] += __shfl_xor(r[v], off, 32);
      }
      if (lm == 0) {
        const int mrow = mt * 16 + lh * 8;
#pragma unroll
        for (int v = 0; v < 8; ++v)
          atomicAdd(&eo[(mrow + v) * E_N + e], r[v]);
      }
    }
  }
  __syncthreads();

  // top-3 (jax tie rule) + softmax + weighted combine
  if (t < MT) {
    const int m = t;
    float g[E_N];
#pragma unroll
    for (int e = 0; e < E_N; ++e) g[e] = gl[m * E_N + e];
    int idx[3];
    float sc[3];
    bool used[E_N] = {false, false, false, false, false, false};
#pragma unroll
    for (int k = 0; k < 3; ++k) {
      float best = -INFINITY;
      int bi = 0;
#pragma unroll
      for (int e = 0; e < E_N; ++e)
        if (!used[e] && g[e] > best) {
          best = g[e];
          bi = e;
        }
      used[bi] = true;
      idx[k] = bi;
      sc[k] = best;
    }
    const float w0 = 1.f;
    const float w1 = __expf(sc[1] - sc[0]);
    const float w2s = __expf(sc[2] - sc[0]);
    const float inv = 1.f / (w0 + w1 + w2s);
    float r = 0.f;
    r += (w0 * inv) * (eo[m * E_N + idx[0]] + b2[idx[0]]);
    r += (w1 * inv) * (eo[m * E_N + idx[1]] + b2[idx[1]]);
    r += (w2s * inv) * (eo[m * E_N + idx[2]] + b2[idx[2]]);
    out[b0 + m] = r;
  }
}

// ---------------------------------------------------------------------------
extern "C" void kernel_launch(void* const* d_in, const int* in_sizes, int n_in,
                              void* d_out, int out_size, void* d_ws,
                              size_t ws_size, hipStream_t stream) {
  (void)in_sizes;
  (void)n_in;
  (void)out_size;
  (void)ws_size;
  const float* x = (const float*)d_in[0];
  // d_in[1]=Wq, d_in[2]=Wk unused by the reference forward pass
  const float* Wv = (const float*)d_in[3];
  const float* Wo = (const float*)d_in[4];
  const float* bo = (const float*)d_in[5];
  const float* Wsp = (const float*)d_in[6];
  const float* bsp = (const float*)d_in[7];
  const float* Wg = (const float*)d_in[8];
  const float* bg = (const float*)d_in[9];
  const float* ml = (const float*)d_in[10];
  const float* W1 = (const float*)d_in[11];
  const float* b1 = (const float*)d_in[12];
  const float* W2 = (const float*)d_in[13];
  const float* b2 = (const float*)d_in[14];
  float* ws = (float*)d_ws;
  float* out = (float*)d_out;

  hipLaunchKernelGGL(moe_setup1, dim3(1), dim3(1024), 0, stream, Wv, Wo, bo,
                     Wsp, bsp, ml, ws);
  hipLaunchKernelGGL(moe_setup2, dim3(E_N * D_DIM), dim3(HID_N), 0, stream, W1,
                     b1, ws);
  hipLaunchKernelGGL(moe_main, dim3(B_TOK / MT), dim3(512), 0, stream, x, Wg,
                     bg, W2, b2, ws, out);
}